// Net_22067541967436
// MI455X (gfx1250) — compile-verified
//
#include <hip/hip_runtime.h>

typedef __attribute__((ext_vector_type(2))) float v2f;
typedef __attribute__((ext_vector_type(8))) float v8f;

#define N_OSC 50
#define N_PER 40
#define STEPS 100
#define DT 0.01f

// ---------------------------------------------------------------------------
// h = relu(x @ fc1_w^T + fc1_b)   (B,2)x(128,2) -> (B,128)
// ---------------------------------------------------------------------------
__global__ void fc1_kernel(const float* __restrict__ x, const float* __restrict__ w1,
                           const float* __restrict__ b1, float* __restrict__ h, int B) {
    int idx = blockIdx.x * blockDim.x + threadIdx.x;
    if (idx >= B * 128) return;
    int i = idx >> 7, j = idx & 127;
    float v = fmaf(x[2 * i], w1[2 * j], fmaf(x[2 * i + 1], w1[2 * j + 1], b1[j]));
    h[idx] = v > 0.f ? v : 0.f;
}

// ---------------------------------------------------------------------------
// direct[i] = h[i,:] . fcd_w + fcd_b       one wave32 per batch element
// ---------------------------------------------------------------------------
__global__ void direct_kernel(const float* __restrict__ h, const float* __restrict__ fcd_w,
                              const float* __restrict__ fcd_b, float* __restrict__ direct, int B) {
    int wave = blockIdx.x * (blockDim.x >> 5) + (threadIdx.x >> 5);
    int lane = threadIdx.x & 31;
    if (wave >= B) return;
    const float* hr = h + (size_t)wave * 128;
    float s = 0.f;
    #pragma unroll
    for (int j = lane; j < 128; j += 32) s = fmaf(hr[j], fcd_w[j], s);
    #pragma unroll
    for (int off = 16; off > 0; off >>= 1) s += __shfl_xor(s, off, 32);
    if (lane == 0) direct[wave] = s + fcd_b[0];
}

// ---------------------------------------------------------------------------
// C[M,N] = act(A[M,K] @ W[N,K]^T + bias[N])  via V_WMMA_F32_16X16X4_F32
// One wave per 16x16 tile; K marched in steps of 4 (full fp32 accumulate).
// A frag (16x4 f32, 2 VGPRs): lanes 0-15 -> M=lane, K={0,1}; lanes 16-31 -> K={2,3}
// B frag (4x16 f32, 2 VGPRs): lanes 0-15 -> N=lane, K={0,1}; lanes 16-31 -> K={2,3}
// C/D (8 VGPRs): VGPR v, lane l -> M = v + 8*(l>>4), N = l&15
// ---------------------------------------------------------------------------
__global__ void gemm_wmma_f32(const float* __restrict__ A, const float* __restrict__ W,
                              const float* __restrict__ bias, float* __restrict__ C,
                              int M, int N, int K, int ldc, int relu) {
    int wavesPerBlock = blockDim.x >> 5;
    int waveId = blockIdx.x * wavesPerBlock + (threadIdx.x >> 5);
    int ntN = (N + 15) >> 4;
    int ntM = M >> 4;
    if (waveId >= ntM * ntN) return;          // wave-uniform: EXEC stays all-1s for WMMA
    int mt = waveId / ntN;
    int nt = waveId - mt * ntN;
    int lane = threadIdx.x & 31;
    int lm   = lane & 15;
    int half = lane >> 4;                     // 0: K={0,1}, 1: K={2,3}
    int m0 = mt << 4, n0 = nt << 4;

    const float* arow = A + (size_t)(m0 + lm) * K + 2 * half;
    int wr = n0 + lm; if (wr > N - 1) wr = N - 1;   // clamp OOB rows (fc3: N=100)
    const float* wrow = W + (size_t)wr * K + 2 * half;

    v8f acc = {};
    for (int k0 = 0; k0 < K; k0 += 4) {
        v2f a = *(const v2f*)(arow + k0);     // 8B aligned: K=128, even offsets
        v2f b = *(const v2f*)(wrow + k0);
        acc = __builtin_amdgcn_wmma_f32_16x16x4_f32(
            false, a, false, b, (short)0, acc, false, false);
    }

    int ccol = n0 + lm;
    if (ccol < N) {
        float bv = bias[ccol];
        int crow = m0 + 8 * half;
        #pragma unroll
        for (int v = 0; v < 8; ++v) {
            float val = acc[v] + bv;
            if (relu) val = val > 0.f ? val : 0.f;
            C[(size_t)(crow + v) * ldc + ccol] = val;
        }
    }
}

// ---------------------------------------------------------------------------
// 100-step oscillator + pendulum scan. One thread per batch element.
// o-state (100 floats/elem) lives in LDS, column layout -> conflict-free.
// Weight tables indexed only by loop counters -> uniform scalar-cache loads.
// ---------------------------------------------------------------------------
__global__ __launch_bounds__(128)
void scan_kernel(const float* __restrict__ x,        // (B,2)
                 const float* __restrict__ x3,       // (B,100)
                 const float* __restrict__ direct,   // (B)
                 const float* __restrict__ enc,      // (50,40,2)
                 const float* __restrict__ obias,    // (50,40)
                 const float* __restrict__ dec,      // (50,2,40)
                 const float* __restrict__ fc4_w,    // (2000)
                 const float* __restrict__ fc4_b,    // (1)
                 float* __restrict__ out, int B) {
    __shared__ float s_o[2 * N_OSC * 128];           // 51.2 KB
    int tid = threadIdx.x;
    int i = blockIdx.x * 128 + tid;

    #pragma unroll 4
    for (int k = 0; k < 2 * N_OSC; ++k)
        s_o[k * 128 + tid] = x3[(size_t)i * 100 + k];   // k = osc*2 + d

    float theta = x[2 * i];
    float omega = 0.f;
    float dir = direct[i] + fc4_b[0];

    const size_t tq_off = (size_t)STEPS * B * 2;

    for (int s = 0; s < STEPS; ++s) {
        float tq = dir;
        for (int osc = 0; osc < N_OSC; ++osc) {
            float o0 = s_o[(2 * osc) * 128 + tid];
            float o1 = s_o[(2 * osc + 1) * 128 + tid];
            float d0 = 0.f, d1 = 0.f;
            const float* e  = enc   + osc * 80;
            const float* ob = obias + osc * 40;
            const float* dc = dec   + osc * 80;
            const float* f4 = fc4_w + osc * 40;
            #pragma unroll 8
            for (int n = 0; n < N_PER; ++n) {
                float a = fmaf(o0, e[2 * n], fmaf(o1, e[2 * n + 1], ob[n]));
                a = a > 0.f ? a : 0.f;
                d0 = fmaf(dc[n],      a, d0);
                d1 = fmaf(dc[40 + n], a, d1);
                tq = fmaf(f4[n],      a, tq);
            }
            s_o[(2 * osc) * 128 + tid]     = fmaf(DT, d0, o0);
            s_o[(2 * osc + 1) * 128 + tid] = fmaf(DT, d1, o1);
        }
        float alpha = tq - sinf(theta) - 0.1f * omega;  // M_G_LC=1, I=1
        float nth = fmaf(DT, omega, theta);
        float nom = fmaf(DT, alpha, omega);
        theta = nth; omega = nom;
        out[(size_t)s * B * 2 + (size_t)i * 2 + 0] = theta;   // l_states
        out[(size_t)s * B * 2 + (size_t)i * 2 + 1] = omega;
        out[tq_off + (size_t)s * B + i] = tq;                 // torques
    }
}

// ---------------------------------------------------------------------------
extern "C" void kernel_launch(void* const* d_in, const int* in_sizes, int n_in,
                              void* d_out, int out_size, void* d_ws, size_t ws_size,
                              hipStream_t stream) {
    const float* x     = (const float*)d_in[0];
    const float* fc1_w = (const float*)d_in[1];
    const float* fc1_b = (const float*)d_in[2];
    const float* fc2_w = (const float*)d_in[3];
    const float* fc2_b = (const float*)d_in[4];
    const float* fc3_w = (const float*)d_in[5];
    const float* fc3_b = (const float*)d_in[6];
    const float* fcd_w = (const float*)d_in[7];
    const float* fcd_b = (const float*)d_in[8];
    const float* enc   = (const float*)d_in[9];
    const float* obias = (const float*)d_in[10];
    const float* dec   = (const float*)d_in[11];
    const float* fc4_w = (const float*)d_in[12];
    const float* fc4_b = (const float*)d_in[13];

    const int B = in_sizes[0] / 2;            // 8192

    float* ws      = (float*)d_ws;
    float* h       = ws;                      // B*128
    float* h2      = h  + (size_t)B * 128;    // B*128
    float* x3      = h2 + (size_t)B * 128;    // B*100
    float* direct  = x3 + (size_t)B * 100;    // B

    // fc1 + relu
    fc1_kernel<<<(B * 128 + 255) / 256, 256, 0, stream>>>(x, fc1_w, fc1_b, h, B);

    // direct = h @ fcd_w^T + fcd_b
    direct_kernel<<<(B + 7) / 8, 256, 0, stream>>>(h, fcd_w, fcd_b, direct, B);

    // h2 = relu(h @ fc2_w^T + fc2_b)   — WMMA fp32
    {
        int tiles = (B / 16) * ((128 + 15) / 16);
        gemm_wmma_f32<<<(tiles + 7) / 8, 256, 0, stream>>>(
            h, fc2_w, fc2_b, h2, B, 128, 128, 128, 1);
    }
    // x3 = h2 @ fc3_w^T + fc3_b        — WMMA fp32, N=100 masked
    {
        int tiles = (B / 16) * ((100 + 15) / 16);
        gemm_wmma_f32<<<(tiles + 7) / 8, 256, 0, stream>>>(
            h2, fc3_w, fc3_b, x3, B, 100, 128, 100, 0);
    }

    // 100-step scan -> d_out = [l_states (100,B,2) | torques (100,B)]
    scan_kernel<<<B / 128, 128, 0, stream>>>(
        x, x3, direct, enc, obias, dec, fc4_w, fc4_b, (float*)d_out, B);
}